// PositionalSelfAttentionClassifier_40157944218160
// MI455X (gfx1250) — compile-verified
//
#include <hip/hip_runtime.h>

typedef float v2f __attribute__((ext_vector_type(2)));
typedef float v8f __attribute__((ext_vector_type(8)));

#define BATCH  32
#define SEQ    2048
#define DIM    512
#define NCLS   4
#define TTILE  128
#define NTILE  (SEQ / TTILE)   // 16

// ---------------------------------------------------------------------------
// Kernel 1: scores[b,t] = q_b . (emb_table[x[b,t]] + pos_table[t])
// Uses V_WMMA_F32_16X16X4_F32: A = q broadcast into all 16 rows,
// B columns = 16 embedding rows; accumulate over K in chunks of 4.
// D row M=0 (acc[0], lanes 0-15, N=lane) holds the 16 scores.
// ---------------------------------------------------------------------------
__global__ __launch_bounds__(256) void scores_kernel(
    const int* __restrict__ x, const float* __restrict__ emb_table,
    const float* __restrict__ pos_table, float* __restrict__ scores)
{
  __shared__ float q[DIM];
  __shared__ int   toks[TTILE];
  const int tile = blockIdx.x;
  const int b    = blockIdx.y;
  const int tid  = threadIdx.x;

  // q_b = emb_table[x[b,0]] + pos_table[0]
  const int tok0 = x[b * SEQ];
  for (int d = tid; d < DIM; d += 256)
    q[d] = emb_table[(size_t)tok0 * DIM + d] + pos_table[d];
  for (int i = tid; i < TTILE; i += 256)
    toks[i] = x[b * SEQ + tile * TTILE + i];
  __syncthreads();

  const int wave  = tid >> 5;
  const int lane  = tid & 31;
  const int n     = lane & 15;   // B column / output N
  const int khalf = lane >> 4;   // which K pair this half-wave holds
  const int t0    = tile * TTILE + wave * 16;

  // Per ISA layout for 32-bit A(16x4)/B(4x16):
  //  lanes 0-15 hold K={k0,k0+1}, lanes 16-31 hold K={k0+2,k0+3}
  const v2f* er = (const v2f*)(emb_table + (size_t)toks[wave * 16 + n] * DIM);
  const v2f* pr = (const v2f*)(pos_table + (size_t)(t0 + n) * DIM);
  const v2f* q2 = (const v2f*)q;

  v8f acc = {};
  #pragma unroll 4
  for (int k0 = 0; k0 < DIM; k0 += 4) {
    const int i2 = (k0 >> 1) + khalf;        // float2 index -> cols kk, kk+1
    v2f bm = er[i2] + pr[i2];                // B[k][n] = emb row n, cols kk..kk+1
    v2f am = q2[i2];                         // A[m][k] = q[k] (same for all m)
    acc = __builtin_amdgcn_wmma_f32_16x16x4_f32(
        /*neg_a=*/false, am, /*neg_b=*/false, bm,
        /*c_mod=*/(short)0, acc, /*reuse_a=*/false, /*reuse_b=*/false);
  }
  // D[M=0][N=lane] lives in acc[0] on lanes 0..15
  if (lane < 16)
    scores[(size_t)b * SEQ + t0 + lane] = acc[0];
}

// ---------------------------------------------------------------------------
// Kernel 2: per-batch softmax stats (max, sum of exp)
// ---------------------------------------------------------------------------
__global__ __launch_bounds__(256) void stats_kernel(
    const float* __restrict__ scores, float* __restrict__ stats)
{
  __shared__ float red[256];
  const int b   = blockIdx.x;
  const int tid = threadIdx.x;
  const float* s = scores + (size_t)b * SEQ;

  float m = -3.402823466e38f;
  for (int t = tid; t < SEQ; t += 256) m = fmaxf(m, s[t]);
  red[tid] = m; __syncthreads();
  for (int o = 128; o > 0; o >>= 1) {
    if (tid < o) red[tid] = fmaxf(red[tid], red[tid + o]);
    __syncthreads();
  }
  m = red[0]; __syncthreads();

  float z = 0.f;
  for (int t = tid; t < SEQ; t += 256) z += expf(s[t] - m);
  red[tid] = z; __syncthreads();
  for (int o = 128; o > 0; o >>= 1) {
    if (tid < o) red[tid] += red[tid + o];
    __syncthreads();
  }
  if (tid == 0) { stats[b * 2] = m; stats[b * 2 + 1] = red[0]; }
}

// ---------------------------------------------------------------------------
// Kernel 3: partial weighted row-sums:
//   ypart[b,chunk,d] = sum_{t in chunk} softmax_w[t] * emb[b,t,d]
// Block reads one full 2KB emb row per iteration, fully coalesced.
// ---------------------------------------------------------------------------
__global__ __launch_bounds__(256) void ypartial_kernel(
    const int* __restrict__ x, const float* __restrict__ emb_table,
    const float* __restrict__ pos_table, const float* __restrict__ scores,
    const float* __restrict__ stats, float* __restrict__ ypart)
{
  __shared__ float w[TTILE];
  __shared__ int   toks[TTILE];
  const int chunk = blockIdx.x;
  const int b     = blockIdx.y;
  const int tid   = threadIdx.x;

  const float m    = stats[b * 2];
  const float zinv = 1.f / stats[b * 2 + 1];
  for (int i = tid; i < TTILE; i += 256) {
    const int t = chunk * TTILE + i;
    w[i]    = expf(scores[(size_t)b * SEQ + t] - m) * zinv;
    toks[i] = x[b * SEQ + t];
  }
  __syncthreads();

  float ax = 0.f, ay = 0.f;       // this thread owns columns 2*tid, 2*tid+1
  for (int i = 0; i < TTILE; ++i) {
    const int t = chunk * TTILE + i;
    const float2 e = ((const float2*)(emb_table + (size_t)toks[i] * DIM))[tid];
    const float2 p = ((const float2*)(pos_table + (size_t)t * DIM))[tid];
    const float wt = w[i];
    ax = fmaf(wt, e.x + p.x, ax);
    ay = fmaf(wt, e.y + p.y, ay);
  }
  float2* out = (float2*)(ypart + (size_t)(b * NTILE + chunk) * DIM);
  out[tid] = make_float2(ax, ay);
}

// ---------------------------------------------------------------------------
// Kernel 4: reduce partials -> y0, then logits = y0 @ fc_w.T + fc_b
// ---------------------------------------------------------------------------
__global__ __launch_bounds__(256) void logits_kernel(
    const float* __restrict__ ypart, const float* __restrict__ fc_w,
    const float* __restrict__ fc_b, float* __restrict__ out)
{
  __shared__ float red[256];
  const int b   = blockIdx.x;
  const int tid = threadIdx.x;

  float yx = 0.f, yy = 0.f;
  for (int c = 0; c < NTILE; ++c) {
    const float2 v = ((const float2*)(ypart + (size_t)(b * NTILE + c) * DIM))[tid];
    yx += v.x; yy += v.y;
  }
  for (int cls = 0; cls < NCLS; ++cls) {
    float p = yx * fc_w[cls * DIM + 2 * tid] + yy * fc_w[cls * DIM + 2 * tid + 1];
    red[tid] = p; __syncthreads();
    for (int o = 128; o > 0; o >>= 1) {
      if (tid < o) red[tid] += red[tid + o];
      __syncthreads();
    }
    if (tid == 0) out[b * NCLS + cls] = red[0] + fc_b[cls];
    __syncthreads();
  }
}

// ---------------------------------------------------------------------------
extern "C" void kernel_launch(void* const* d_in, const int* in_sizes, int n_in,
                              void* d_out, int out_size, void* d_ws, size_t ws_size,
                              hipStream_t stream) {
  (void)in_sizes; (void)n_in; (void)out_size; (void)ws_size;
  const int*   x         = (const int*)d_in[0];
  const float* emb_table = (const float*)d_in[1];
  const float* pos_table = (const float*)d_in[2];
  const float* fc_w      = (const float*)d_in[3];
  const float* fc_b      = (const float*)d_in[4];
  float* out = (float*)d_out;

  float* ws     = (float*)d_ws;
  float* scores = ws;                         // 32*2048 floats (256 KB)
  float* stats  = ws + BATCH * SEQ;           // 64 floats
  float* ypart  = ws + BATCH * SEQ + 512;     // 32*16*512 floats (1 MB)

  scores_kernel  <<<dim3(NTILE, BATCH), 256, 0, stream>>>(x, emb_table, pos_table, scores);
  stats_kernel   <<<BATCH,             256, 0, stream>>>(scores, stats);
  ypartial_kernel<<<dim3(NTILE, BATCH), 256, 0, stream>>>(x, emb_table, pos_table, scores, stats, ypart);
  logits_kernel  <<<BATCH,             256, 0, stream>>>(ypart, fc_w, fc_b, out);
}